// MoleculeModel_17154099380405
// MI455X (gfx1250) — compile-verified
//
#include <hip/hip_runtime.h>
#include <stdint.h>

// ---------------------------------------------------------------------------
// Types
// ---------------------------------------------------------------------------
typedef float v2f __attribute__((ext_vector_type(2)));
typedef float v8f __attribute__((ext_vector_type(8)));
typedef unsigned int tdm_uv4 __attribute__((ext_vector_type(4)));
typedef int          tdm_iv8 __attribute__((ext_vector_type(8)));
typedef int          tdm_iv4 __attribute__((ext_vector_type(4)));

#define USE_TDM 1

// ---------------------------------------------------------------------------
// TDM 2D tile load: global (strided rows) -> LDS (packed), fp32 elements.
// Builds D# groups per CDNA5 ISA §8 (group0 128b, group1 256b, groups2/3 zero
// for a 2D tensor). data_size code 2 == 4 bytes. One issue per wave.
// 6-arg builtin: (uint32x4 g0, int32x8 g1, int32x4 g2, int32x4 g3,
//                 int32x8 extra, i32 cpol)
// ---------------------------------------------------------------------------
__device__ __forceinline__ void tdm_load_tile_f32(uint32_t lds_byte_addr,
                                                  const float* gptr,
                                                  uint32_t tile_w,   // elems per row
                                                  uint32_t tile_h,   // rows
                                                  uint32_t row_stride_elems) {
#if USE_TDM && __has_builtin(__builtin_amdgcn_tensor_load_to_lds)
  uint64_t ga = (uint64_t)(uintptr_t)gptr;
  tdm_uv4 g0;
  g0.x = 1u;                                       // count=1, user mode
  g0.y = lds_byte_addr;                            // lds_addr
  g0.z = (uint32_t)(ga & 0xFFFFFFFFu);             // global_addr[31:0]
  g0.w = (uint32_t)((ga >> 32) & 0x1FFFFFFu) | (2u << 30); // addr[56:32] | type=2
  tdm_iv8 g1;
  g1[0] = (int)(2u << 16);                               // wg_mask=0, data_size=4B
  g1[1] = (int)((tile_w & 0xFFFFu) << 16);               // tensor_dim0 lo16
  g1[2] = (int)(((tile_w >> 16) & 0xFFFFu) | ((tile_h & 0xFFFFu) << 16)); // dim0 hi | dim1 lo
  g1[3] = (int)(((tile_h >> 16) & 0xFFFFu) | ((tile_w & 0xFFFFu) << 16)); // dim1 hi | tile_dim0
  g1[4] = (int)(tile_h & 0xFFFFu);                       // tile_dim1, tile_dim2=0
  g1[5] = (int)row_stride_elems;                         // tensor_dim0_stride lo32
  g1[6] = 0;
  g1[7] = 0;
  tdm_iv4 z4 = (tdm_iv4)0;
  tdm_iv8 z8 = (tdm_iv8)0;
  __builtin_amdgcn_tensor_load_to_lds(g0, g1, z4, z4, z8, 0);
  __builtin_amdgcn_s_wait_tensorcnt(0);
#else
  float* lds = (float*)(uintptr_t)lds_byte_addr; (void)lds;
  (void)gptr; (void)tile_w; (void)tile_h; (void)row_stride_elems;
#endif
}

// ---------------------------------------------------------------------------
// Generic fp32 WMMA GEMM: C[M,N] = op(A[M,K] x B) (+bias) (ReLU)
//  TRANSB=0: B is [K,N] row-major;  TRANSB=1: B is [N,K] row-major (B^T used)
//  NGUARD=1: N not a multiple of 32 -> bounds-checked B loads / stores.
//  Block: 256 threads = 8 waves, block tile 64x32, K-chunk 32.
//  Each wave computes one 16x16 tile with v_wmma_f32_16x16x4_f32.
//  Pipelining: global loads for chunk k are issued before the barrier that
//  protects the LDS overwrite (hidden under chunk k-1 compute); all 16 LDS
//  fragments are preloaded into registers before the 8-WMMA chain.
// ---------------------------------------------------------------------------
template <bool TRANSB, bool BIAS, bool RELU, bool NGUARD>
__global__ __launch_bounds__(256) void gemm_wmma_f32(
    const float* __restrict__ A, const float* __restrict__ B,
    const float* __restrict__ bias, float* __restrict__ C,
    int M, int N, int K) {
  const int t = threadIdx.x;
  const int n0 = blockIdx.x * 32;
  const int m0 = blockIdx.y * 64;

  __shared__ float As[64][33];
  __shared__ float Bs[32][33];

  const int wv = t >> 5;
  const int l = t & 31;
  const int hi = l >> 4;     // selects K pair / M half per WMMA layout
  const int lo = l & 15;
  const int m_off = (wv & 3) * 16;
  const int n_off = (wv >> 2) * 16;

  // Staging geometry (constant per thread)
  const int rowA = t >> 3;            // 0..31 (second float4 at rowA+32)
  const int c4A = (t & 7) * 4;        // col within 32-wide K chunk
  const float* aPtr0 = A + (size_t)(m0 + rowA) * K + c4A;
  const float* aPtr1 = A + (size_t)(m0 + rowA + 32) * K + c4A;

  const int rowB = t >> 3;            // !TRANSB: k in chunk; TRANSB: n in tile
  const int cbB = (t & 7) * 4;        // !TRANSB: n base;     TRANSB: k base
  const float* bPtr = TRANSB ? (B + (size_t)(n0 + rowB) * K + cbB)
                             : (B + (size_t)rowB * N + n0 + cbB);
  const size_t bStep = TRANSB ? 32 : (size_t)32 * N;

  v8f acc = {0.f, 0.f, 0.f, 0.f, 0.f, 0.f, 0.f, 0.f};

  for (int kc = 0; kc < K; kc += 32) {
    // Issue global loads for this chunk BEFORE the LDS-overwrite barrier.
    const float4 a0 = *(const float4*)aPtr0;
    const float4 a1 = *(const float4*)aPtr1;
    float4 b0;
    if (!NGUARD || TRANSB) b0 = *(const float4*)bPtr;
    aPtr0 += 32; aPtr1 += 32;

    __syncthreads();   // previous chunk's LDS reads complete

    As[rowA][c4A + 0] = a0.x; As[rowA][c4A + 1] = a0.y;
    As[rowA][c4A + 2] = a0.z; As[rowA][c4A + 3] = a0.w;
    As[rowA + 32][c4A + 0] = a1.x; As[rowA + 32][c4A + 1] = a1.y;
    As[rowA + 32][c4A + 2] = a1.z; As[rowA + 32][c4A + 3] = a1.w;

    if (!TRANSB) {
      if (!NGUARD) {
        Bs[rowB][cbB + 0] = b0.x; Bs[rowB][cbB + 1] = b0.y;
        Bs[rowB][cbB + 2] = b0.z; Bs[rowB][cbB + 3] = b0.w;
      } else {
#pragma unroll
        for (int j = 0; j < 4; ++j) {
          const int col = n0 + cbB + j;
          Bs[rowB][cbB + j] = (col < N) ? B[(size_t)(kc + rowB) * N + col] : 0.f;
        }
      }
    } else {
      Bs[cbB + 0][rowB] = b0.x; Bs[cbB + 1][rowB] = b0.y;
      Bs[cbB + 2][rowB] = b0.z; Bs[cbB + 3][rowB] = b0.w;
    }
    bPtr += bStep;
    __syncthreads();

    // Preload all fragments, then run the WMMA chain back-to-back.
    v2f af[8], bf[8];
#pragma unroll
    for (int i = 0; i < 8; ++i) {
      af[i].x = As[m_off + lo][4 * i + 2 * hi];
      af[i].y = As[m_off + lo][4 * i + 2 * hi + 1];
      bf[i].x = Bs[4 * i + 2 * hi][n_off + lo];
      bf[i].y = Bs[4 * i + 2 * hi + 1][n_off + lo];
    }
#pragma unroll
    for (int i = 0; i < 8; ++i) {
      acc = __builtin_amdgcn_wmma_f32_16x16x4_f32(
          /*neg_a=*/false, af[i], /*neg_b=*/false, bf[i],
          /*c_mod=*/(short)0, acc, /*reuse_a=*/false, /*reuse_b=*/false);
    }
  }

  // Epilogue: lane holds N=lo(+n_off), rows M = v + 8*hi (16x16 f32 C layout)
  const int col = n0 + n_off + lo;
  if (!NGUARD || col < N) {
    const float bv = BIAS ? bias[col] : 0.f;
#pragma unroll
    for (int v = 0; v < 8; ++v) {
      const int m = m0 + m_off + v + 8 * hi;
      float x = acc[v] + bv;
      if (RELU) x = fmaxf(x, 0.f);
      C[(size_t)m * N + col] = x;
    }
  }
  (void)M;
}

// ---------------------------------------------------------------------------
// Fused per-segment kernel (one block per graph b, 256 threads):
//   scores[a] = atoms[b*64+a] . Q[b]        (segment constant cancels in softmax)
//   w = softmax_64(scores); write scores output
//   pooled[h] = sum_a w[a]*atoms[b*64+a][h]
//   h = pooled * partner[b];  pert = h + sign(h)*noise[b]/||noise[b]||*0.1
//   write h/pert (if WRITE_H) and pert into pair buffer column block.
// Atom tiles (64x128 fp32, row stride 512) are DMA'd to LDS by the TDM.
// ---------------------------------------------------------------------------
template <bool WRITE_H>
__global__ __launch_bounds__(256) void scores_pool(
    const float* __restrict__ atoms,     // [B*64, 512]
    const float* __restrict__ Q,         // [B, 512]
    const float* __restrict__ partner,   // [B, 512]
    const float* __restrict__ noise,     // [B, 512]
    float* __restrict__ scores_out,      // [B*64]
    float* __restrict__ h_out,           // [B,512] or null
    float* __restrict__ h_pert_out,      // [B,512] or null
    float* __restrict__ pair_out,        // [B,1024]
    int pair_off) {
  const int t = threadIdx.x;
  const int b = blockIdx.x;

  __shared__ float tile[64 * 128];   // 32 KB DMA target
  __shared__ float qv[512];
  __shared__ float sc[64];
  __shared__ float red[256];
  __shared__ float snorm_s, zinv_s;

  const uint32_t tile_lds = (uint32_t)(uintptr_t)(&tile[0]);
  const float* seg = atoms + (size_t)b * 64 * 512;

  // Phase 0: stage Q[b], noise norm, zero scores
  qv[t] = Q[(size_t)b * 512 + t];
  qv[t + 256] = Q[(size_t)b * 512 + 256 + t];
  {
    const float n0 = noise[(size_t)b * 512 + t];
    const float n1 = noise[(size_t)b * 512 + 256 + t];
    red[t] = n0 * n0 + n1 * n1;
  }
  if (t < 64) sc[t] = 0.f;
  __syncthreads();
  for (int off = 128; off > 0; off >>= 1) {
    if (t < off) red[t] += red[t + off];
    __syncthreads();
  }
  if (t == 0) snorm_s = 0.1f / fmaxf(sqrtf(red[0]), 1e-12f);

  // Phase 1: scores, 4 slabs of 128 columns
  for (int s = 0; s < 4; ++s) {
    __syncthreads();
    if (t < 32) tdm_load_tile_f32(tile_lds, seg + s * 128, 128, 64, 512);
    __syncthreads();
    const int a = t >> 2;
    const int q4 = t & 3;
    const float* rowp = &tile[a * 128 + q4 * 32];
    const float* qp = &qv[s * 128 + q4 * 32];
    float p = 0.f;
#pragma unroll
    for (int j = 0; j < 32; ++j) p += rowp[j] * qp[j];
    red[t] = p;
    __syncthreads();
    if (t < 64) sc[t] += red[4 * t] + red[4 * t + 1] + red[4 * t + 2] + red[4 * t + 3];
  }
  __syncthreads();

  // Phase 2: softmax over 64 (serial by t0 — 64 elems, deterministic)
  if (t == 0) {
    float m = -3.402823466e+38f;
    for (int a = 0; a < 64; ++a) m = fmaxf(m, sc[a]);
    float z = 0.f;
    for (int a = 0; a < 64; ++a) {
      const float e = expf(sc[a] - m);
      sc[a] = e;
      z += e;
    }
    zinv_s = 1.f / z;
  }
  __syncthreads();
  if (t < 64) {
    const float w = sc[t] * zinv_s;
    sc[t] = w;
    scores_out[(size_t)b * 64 + t] = w;
  }
  __syncthreads();

  // Phase 3: weighted pool (slabs re-DMA'd; L2-resident), epilogue math
  for (int s = 0; s < 4; ++s) {
    __syncthreads();
    if (t < 32) tdm_load_tile_f32(tile_lds, seg + s * 128, 128, 64, 512);
    __syncthreads();
    const int c = t & 127;
    const int half = t >> 7;
    float p = 0.f;
#pragma unroll
    for (int a = 0; a < 32; ++a) p += sc[half * 32 + a] * tile[(half * 32 + a) * 128 + c];
    red[t] = p;
    __syncthreads();
    if (t < 128) {
      const int col = s * 128 + t;
      const float pooled = red[t] + red[t + 128];
      const float h = pooled * partner[(size_t)b * 512 + col];
      const float sg = (h > 0.f) ? 1.f : ((h < 0.f) ? -1.f : 0.f);
      const float pert = h + sg * noise[(size_t)b * 512 + col] * snorm_s;
      if (WRITE_H) {
        h_out[(size_t)b * 512 + col] = h;
        h_pert_out[(size_t)b * 512 + col] = pert;
      }
      pair_out[(size_t)b * 1024 + pair_off + col] = pert;
    }
  }
}

// ---------------------------------------------------------------------------
// Host launcher
// ---------------------------------------------------------------------------
extern "C" void kernel_launch(void* const* d_in, const int* in_sizes, int n_in,
                              void* d_out, int out_size, void* d_ws, size_t ws_size,
                              hipStream_t stream) {
  (void)in_sizes; (void)n_in; (void)out_size; (void)ws_size;

  const float* left_output  = (const float*)d_in[0];   // [1024,512]
  const float* right_output = (const float*)d_in[1];   // [1024,512]
  const float* left_atom    = (const float*)d_in[2];   // [65536,512]
  const float* right_atom   = (const float*)d_in[3];   // [65536,512]
  const float* noise        = (const float*)d_in[6];   // [1024,512]
  const float* w_i_w        = (const float*)d_in[7];   // [512,512]
  // d_in[8] w_i_b: segment-constant in softmax -> cancels exactly; unused.
  const float* prj_i_w      = (const float*)d_in[9];   // [512,512]
  const float* prj_i_b      = (const float*)d_in[10];  // [512]
  const float* ffn1_w       = (const float*)d_in[11];  // [1024,2048]
  const float* ffn1_b       = (const float*)d_in[12];  // [2048]
  const float* ffn2_w       = (const float*)d_in[13];  // [2048,86]
  const float* ffn2_b       = (const float*)d_in[14];  // [86]

  float* out = (float*)d_out;
  float* out_logits = out;                 // [1024,86]
  float* out_h      = out + 88064;         // [1024,512]
  float* out_hp     = out + 612352;        // [1024,512]
  float* out_ls     = out + 1136640;       // [65536]
  float* out_rs     = out + 1202176;       // [65536]
  float* out_lo     = out + 1267712;       // [1024,512]
  float* out_ro     = out + 1792000;       // [1024,512]

  float* ws   = (float*)d_ws;
  float* P1   = ws;                        // 1024*512
  float* P2   = P1 + 524288;
  float* Q1   = P2 + 524288;
  float* Q2   = Q1 + 524288;
  float* pair = Q2 + 524288;               // 1024*1024
  float* hid  = pair + 1048576;            // 1024*2048

  const dim3 blk(256);

  // P = output @ prj_w + prj_b   (left scores use right_output's projection)
  gemm_wmma_f32<false, true,  false, false><<<dim3(16, 16), blk, 0, stream>>>(
      right_output, prj_i_w, prj_i_b, P1, 1024, 512, 512);
  gemm_wmma_f32<false, true,  false, false><<<dim3(16, 16), blk, 0, stream>>>(
      left_output,  prj_i_w, prj_i_b, P2, 1024, 512, 512);
  // Q[b,k] = sum_h P[b,h] * w_i_w[k,h]  (B transposed)
  gemm_wmma_f32<true,  false, false, false><<<dim3(16, 16), blk, 0, stream>>>(
      P1, w_i_w, nullptr, Q1, 1024, 512, 512);
  gemm_wmma_f32<true,  false, false, false><<<dim3(16, 16), blk, 0, stream>>>(
      P2, w_i_w, nullptr, Q2, 1024, 512, 512);

  // Fused scores + softmax + pool + perturb (TDM-staged atom tiles)
  scores_pool<true ><<<dim3(1024), blk, 0, stream>>>(
      left_atom,  Q1, right_output, noise, out_ls, out_h, out_hp, pair, 0);
  scores_pool<false><<<dim3(1024), blk, 0, stream>>>(
      right_atom, Q2, left_output,  noise, out_rs, nullptr, nullptr, pair, 512);

  // FFN
  gemm_wmma_f32<false, true,  true,  false><<<dim3(64, 16), blk, 0, stream>>>(
      pair, ffn1_w, ffn1_b, hid, 1024, 2048, 1024);
  gemm_wmma_f32<false, true,  false, true ><<<dim3(3, 16),  blk, 0, stream>>>(
      hid, ffn2_w, ffn2_b, out_logits, 1024, 86, 2048);

  // Pass-through outputs
  (void)hipMemcpyAsync(out_lo, left_output,  524288 * sizeof(float),
                       hipMemcpyDeviceToDevice, stream);
  (void)hipMemcpyAsync(out_ro, right_output, 524288 * sizeof(float),
                       hipMemcpyDeviceToDevice, stream);
}